// BrainSphereCNN_13726715478120
// MI455X (gfx1250) — compile-verified
//
#include <hip/hip_runtime.h>
#include <hip/hip_bf16.h>

typedef __attribute__((ext_vector_type(2))) float v2f;
typedef __attribute__((ext_vector_type(8))) float v8f;

#define CK   64          // K-chunk staged in LDS per iteration
#define ASTR (CK + 4)    // padded LDS stride for A tile (floats)
#define BSTR (CK + 2)    // padded LDS stride for B tile (floats)

// ---------------------------------------------------------------------------
// Gather-conv GEMM: Y[N,cout] = gather(X,neigh)[N,7*cin] @ W[cout,7*cin]^T + b
// One block = 128 threads = 4 waves. Block tile: 16 rows x 64 cols.
// Per K-chunk: cooperatively stage gathered A (16xCK) and weight B (64xCK)
// into LDS (coalesced, zero-padded), then each wave runs a clean
// ds/ds/v_wmma_f32_16x16x4_f32 inner loop on its 16-col slice.
// ---------------------------------------------------------------------------
__global__ __launch_bounds__(128)
void conv_wmma(const float* __restrict__ X, const int* __restrict__ neigh,
               const float* __restrict__ W, const float* __restrict__ bias,
               float* __restrict__ Y, int N, int cin, int K7, int cout, int sh) {
  __shared__ float As[16 * ASTR];   // 4352 B
  __shared__ float Bs[64 * BSTR];   // 16896 B
  __shared__ int   nbase[16 * 7];

  const int tid  = threadIdx.x;
  const int lane = tid & 31;
  const int wave = tid >> 5;
  const int rowTile = blockIdx.x;               // 16 output rows (nodes)
  const int colBase = blockIdx.y * 64;          // 64 output cols per block
  const int colTile = colBase + wave * 16;      // 16 cols per wave
  const bool active = colTile < cout;

  // Neighbor base offsets for the 16 rows of this tile (clamped; masked at write)
  if (tid < 112) {
    int r = tid / 7, j = tid - r * 7;
    int node = rowTile * 16 + r;
    if (node >= N) node = N - 1;
    nbase[tid] = neigh[node * 7 + j] * cin;
  }

  const int m    = lane & 15;          // row in A frag / col in B frag
  const int koff = (lane >> 4) << 1;   // 0 for lanes 0-15, 2 for lanes 16-31
  const int col  = colTile + m;

  v8f acc = {};
  const unsigned ucin = (unsigned)cin;

  for (int k0 = 0; k0 < K7; k0 += CK) {
    __syncthreads();

    // --- stage gathered A chunk: As[r][kk] = X[neigh[node][k/cin]][k%cin] ---
    for (int idx = tid; idx < 16 * CK; idx += 128) {
      int r  = idx >> 6;              // idx / CK  (CK == 64)
      int kk = idx & (CK - 1);
      int k  = k0 + kk;
      float v = 0.f;
      if (k < K7) {
        unsigned uk = (unsigned)k;
        unsigned j, c;
        if (sh >= 0) { j = uk >> sh; c = uk & (ucin - 1u); }
        else         { j = uk / ucin; c = uk - j * ucin; }
        v = X[(size_t)nbase[r * 7 + (int)j] + (int)c];
      }
      As[r * ASTR + kk] = v;
    }

    // --- stage weight B chunk (coalesced along k): Bs[rw][kk] = W[col][k] ---
    for (int idx = tid; idx < 64 * CK; idx += 128) {
      int rw = idx >> 6;              // local output col 0..63
      int kk = idx & (CK - 1);
      int k  = k0 + kk;
      int wc = colBase + rw;
      if (wc >= cout) wc = cout - 1;  // clamped rows are masked at write
      float v = (k < K7) ? W[(size_t)wc * K7 + k] : 0.f;
      Bs[rw * BSTR + kk] = v;
    }

    __syncthreads();

    if (active) {
      const float* ap = &As[m * ASTR + koff];
      const float* bp = &Bs[(wave * 16 + m) * BSTR + koff];
      #pragma unroll 8
      for (int kk = 0; kk < CK; kk += 4) {
        v2f a = *(const v2f*)(ap + kk);
        v2f b = *(const v2f*)(bp + kk);
        acc = __builtin_amdgcn_wmma_f32_16x16x4_f32(
            false, a, false, b, (short)0, acc, false, false);
      }
    }
  }

  // D layout: VGPR v -> M = v (lanes 0-15) / v+8 (lanes 16-31), N = lane%16
  if (active && col < cout) {
    float bb = bias[col];
    int rbase = rowTile * 16 + ((lane >> 4) << 3);
    #pragma unroll
    for (int v = 0; v < 8; ++v) {
      int row = rbase + v;
      if (row < N) Y[(size_t)row * cout + col] = acc[v] + bb;
    }
  }
}

// ---------------------------------------------------------------------------
// BatchNorm (training stats, biased variance) -> fused scale/shift per channel
// ---------------------------------------------------------------------------
__global__ __launch_bounds__(256)
void bn_stats(const float* __restrict__ X, const float* __restrict__ gamma,
              const float* __restrict__ beta, float* __restrict__ sc,
              int N, int C) {
  __shared__ float s1[256], s2[256];
  int c = blockIdx.x;
  float a = 0.f, b = 0.f;
  for (int n = threadIdx.x; n < N; n += 256) {
    float v = X[(size_t)n * C + c];
    a += v; b += v * v;
  }
  s1[threadIdx.x] = a; s2[threadIdx.x] = b;
  __syncthreads();
  for (int off = 128; off > 0; off >>= 1) {
    if (threadIdx.x < off) {
      s1[threadIdx.x] += s1[threadIdx.x + off];
      s2[threadIdx.x] += s2[threadIdx.x + off];
    }
    __syncthreads();
  }
  if (threadIdx.x == 0) {
    float m   = s1[0] / (float)N;
    float var = s2[0] / (float)N - m * m;
    float scale = gamma[c] * rsqrtf(var + 1e-5f);
    sc[2 * c]     = scale;
    sc[2 * c + 1] = beta[c] - m * scale;
  }
}

__global__ __launch_bounds__(256)
void bn_apply(float* __restrict__ X, const float* __restrict__ sc,
              int total, int C) {
  int t = blockIdx.x * blockDim.x + threadIdx.x;
  if (t >= total) return;
  int c = t % C;
  float v = X[t] * sc[2 * c] + sc[2 * c + 1];
  X[t] = v > 0.f ? v : 0.f;
}

// ---------------------------------------------------------------------------
// Pool: replicates x[neigh[:num*7]].reshape(num, f, 7).mean(2) exactly
// out[i][c] = mean_k X[neigh[i*7 + (c*7+k)/f]][(c*7+k)%f]   (f power of two)
// ---------------------------------------------------------------------------
__global__ __launch_bounds__(256)
void pool7(const float* __restrict__ X, const int* __restrict__ neigh,
           float* __restrict__ Y, int num, int f, int fsh) {
  int t = blockIdx.x * blockDim.x + threadIdx.x;
  if (t >= num * f) return;
  int i = t >> fsh;
  int c = t & (f - 1);
  float s = 0.f;
  #pragma unroll
  for (int k = 0; k < 7; ++k) {
    int tt = c * 7 + k;
    int r  = i * 7 + (tt >> fsh);
    int cc = tt & (f - 1);
    s += X[(size_t)neigh[r] * f + cc];
  }
  Y[t] = s * (1.f / 7.f);
}

// ---------------------------------------------------------------------------
// Upsample (+ optional skip add): out[n<ns] = P[n]; else mean of 2 gathered
// ---------------------------------------------------------------------------
__global__ __launch_bounds__(256)
void up_add(const float* __restrict__ P, const int* __restrict__ ups,
            const float* __restrict__ skip, float* __restrict__ Y,
            int n_small, int n_big, int C) {
  int t = blockIdx.x * blockDim.x + threadIdx.x;
  if (t >= n_big * C) return;
  int n = t / C, c = t - n * C;
  float v;
  if (n < n_small) {
    v = P[(size_t)n * C + c];
  } else {
    int i = n - n_small;
    v = 0.5f * (P[(size_t)ups[2 * i] * C + c] + P[(size_t)ups[2 * i + 1] * C + c]);
  }
  if (skip) v += skip[t];
  Y[t] = v;
}

// ---------------------------------------------------------------------------
extern "C" void kernel_launch(void* const* d_in, const int* in_sizes, int n_in,
                              void* d_out, int out_size, void* d_ws, size_t ws_size,
                              hipStream_t stream) {
  (void)in_sizes; (void)n_in; (void)out_size; (void)ws_size;
  static const int LEV[6] = {10242, 2562, 642, 162, 42, 12};

  // d_in order: x, Ws[15], bs[15], gammas[10], betas[10], neighs[6], ups[5]
  const float* x = (const float*)d_in[0];
  const float* Ws[15]; const float* bs[15];
  for (int i = 0; i < 15; ++i) { Ws[i] = (const float*)d_in[1 + i];  bs[i] = (const float*)d_in[16 + i]; }
  const float* gm[10]; const float* bt[10];
  for (int i = 0; i < 10; ++i) { gm[i] = (const float*)d_in[31 + i]; bt[i] = (const float*)d_in[41 + i]; }
  const int* nb[6]; for (int l = 0; l < 6; ++l) nb[l] = (const int*)d_in[51 + l];
  const int* up[5]; for (int l = 0; l < 5; ++l) up[l] = (const int*)d_in[57 + l];

  // workspace layout (floats)
  float* ws = (float*)d_ws;
  float* F0 = ws;                 // 10242*64 = 655488 cap
  float* F1 = F0 + 655488;
  float* pp = F1 + 655488;
  float* P[5];
  for (int l = 0; l < 5; ++l) { P[l] = pp; pp += LEV[l + 1] * 36; }
  float* U0 = pp; pp += 2562 * 36;
  float* U1 = pp; pp += 2562 * 36;
  float* ST = pp;                 // 2*1024 fused scale/shift

  static const int conv_in[10]  = {3, 64, 64, 128, 128, 256, 256, 512, 512, 1024};
  static const int conv_out[10] = {64, 64, 128, 128, 256, 256, 512, 512, 1024, 1024};

  auto launch_conv = [&](const float* Xp, const int* ng, const float* Wp, const float* bp,
                         float* Yp, int N, int cin, int cout) {
    int K7 = 7 * cin;
    int sh = -1;
    for (int s = 0; s < 12; ++s) if ((1 << s) == cin) { sh = s; break; }
    dim3 grid((unsigned)((N + 15) / 16), (unsigned)((cout + 63) / 64));
    hipLaunchKernelGGL(conv_wmma, grid, dim3(128), 0, stream,
                       Xp, ng, Wp, bp, Yp, N, cin, K7, cout, sh);
  };
  auto launch_bn = [&](float* Xp, const float* g, const float* b, int N, int C) {
    hipLaunchKernelGGL(bn_stats, dim3((unsigned)C), dim3(256), 0, stream, Xp, g, b, ST, N, C);
    int tot = N * C;
    hipLaunchKernelGGL(bn_apply, dim3((unsigned)((tot + 255) / 256)), dim3(256), 0, stream,
                       Xp, ST, tot, C);
  };

  // ----- encoder -----
  float* A = F0; float* B = F1; const float* cur = x;
  for (int lvl = 0; lvl < 5; ++lvl) {
    int N = LEV[lvl], Np = LEV[lvl + 1];
    launch_conv(cur, nb[lvl], Ws[2 * lvl],     bs[2 * lvl],     A, N, conv_in[2 * lvl],     conv_out[2 * lvl]);
    launch_bn(A, gm[2 * lvl], bt[2 * lvl], N, conv_out[2 * lvl]);
    launch_conv(A,   nb[lvl], Ws[2 * lvl + 1], bs[2 * lvl + 1], B, N, conv_in[2 * lvl + 1], conv_out[2 * lvl + 1]);
    launch_bn(B, gm[2 * lvl + 1], bt[2 * lvl + 1], N, conv_out[2 * lvl + 1]);

    int f = conv_out[2 * lvl + 1];
    int fsh = 0; while ((1 << fsh) < f) ++fsh;
    int tot = Np * f;
    hipLaunchKernelGGL(pool7, dim3((unsigned)((tot + 255) / 256)), dim3(256), 0, stream,
                       B, nb[lvl], A, Np, f, fsh);

    launch_conv(A, nb[lvl + 1], Ws[10 + lvl], bs[10 + lvl], P[lvl], Np, f, 36);
    cur = A; float* t = A; A = B; B = t;
  }

  // ----- decoder -----
  const float* p = P[4];
  float* dA = U0; float* dB = U1;
  for (int lvl = 4; lvl >= 1; --lvl) {
    int nsm = LEV[lvl + 1], nbg = LEV[lvl];
    int tot = nbg * 36;
    hipLaunchKernelGGL(up_add, dim3((unsigned)((tot + 255) / 256)), dim3(256), 0, stream,
                       p, up[lvl], P[lvl - 1], dA, nsm, nbg, 36);
    p = dA; float* t = dA; dA = dB; dB = t;
  }
  {
    int tot = LEV[0] * 36;
    hipLaunchKernelGGL(up_add, dim3((unsigned)((tot + 255) / 256)), dim3(256), 0, stream,
                       p, up[0], (const float*)nullptr, (float*)d_out, LEV[1], LEV[0], 36);
  }
}